// MultiScaleQ6_46832323395756
// MI455X (gfx1250) — compile-verified
//
#include <hip/hip_runtime.h>
#include <math.h>

// fp32 WMMA operand types (16x16x4_f32: A=16x4 -> 2 VGPR, B=4x16 -> 2 VGPR, C/D=16x16 -> 8 VGPR)
typedef float v2f __attribute__((ext_vector_type(2)));
typedef float v8f __attribute__((ext_vector_type(8)));

#define D_MODEL 2048
#define WAVES   8
#define BLK_TOK 128   // 8 waves * 16 tokens

__global__ __launch_bounds__(256, 1)
void msq6_fused(const float* __restrict__ x,
                const float* __restrict__ Wp2,
                const float* __restrict__ Wp3,
                const float* __restrict__ Wp6,
                const float* __restrict__ Wo2,
                const float* __restrict__ Wo3,
                const float* __restrict__ Wo6,
                const float* __restrict__ scw,
                float* __restrict__ out,
                int ntok)
{
    // LDS: projection weights (12 rows, row 11 zero-padded), output weights as
    // K-pair interleaved float2 so each lane's B operand is one b64 LDS load,
    // plus per-wave 16x16 staging tiles.  Total = 96KB + 96KB + 8KB + 8KB = 208KB.
    __shared__ __align__(16) float WpL[12 * D_MODEL];
    __shared__ __align__(16) float BoP[6 * D_MODEL * 2];
    __shared__ __align__(16) float projL[WAVES][256];
    __shared__ __align__(16) float eL[WAVES][256];

    const int tid = threadIdx.x;

    // ---- stage weights into LDS (once per block; whole-weight set ~180KB stays in L2 globally)
    for (int i = tid; i < 12 * D_MODEL; i += 256) {
        int r = i >> 11;
        float v;
        if (r < 2)       v = Wp2[i];
        else if (r < 5)  v = Wp3[i - 2 * D_MODEL];
        else if (r < 11) v = Wp6[i - 5 * D_MODEL];
        else             v = 0.0f;           // zero pad row 11 (N pad 11->12)
        WpL[i] = v;
    }
    for (int i = tid; i < 6 * D_MODEL; i += 256) {
        int kk = i >> 11;                    // K-pair index 0..5
        int dd = i & (D_MODEL - 1);
        int K0 = 2 * kk, K1 = K0 + 1;
        float b0 = (K0 < 2) ? Wo2[dd * 2 + K0]
                 : (K0 < 5) ? Wo3[dd * 3 + K0 - 2]
                 : (K0 < 11) ? Wo6[dd * 6 + K0 - 5] : 0.0f;
        float b1 = (K1 < 2) ? Wo2[dd * 2 + K1]
                 : (K1 < 5) ? Wo3[dd * 3 + K1 - 2]
                 : (K1 < 11) ? Wo6[dd * 6 + K1 - 5] : 0.0f;
        BoP[2 * i]     = b0;
        BoP[2 * i + 1] = b1;
    }
    __syncthreads();

    const int wave = tid >> 5;
    const int lane = tid & 31;
    const int hv   = lane >> 4;   // which 16-lane half
    const int lid  = lane & 15;
    const long long tok0 = (long long)blockIdx.x * BLK_TOK + wave * 16;

    // =========== stage 1: proj(16x12) = x(16x2048) @ Wp^T  via V_WMMA_F32_16X16X4_F32 ==========
    // A layout: lane = hv*16 + m holds A[m][kb + 2*hv + {0,1}]  (ISA 32-bit A 16x4 table)
    // B layout: lane holds B[kb + 2*hv + {0,1}][n = lid]
    v8f acc = {0.f, 0.f, 0.f, 0.f, 0.f, 0.f, 0.f, 0.f};
    const float* xa = x + (tok0 + lid) * D_MODEL + 2 * hv;
    const float* ba = &WpL[(lid < 11 ? lid : 11) * D_MODEL + 2 * hv];
    #pragma unroll 8
    for (int kb = 0; kb < D_MODEL; kb += 4) {
        v2f a = *(const v2f*)(xa + kb);
        v2f b = *(const v2f*)(ba + kb);
        acc = __builtin_amdgcn_wmma_f32_16x16x4_f32(false, a, false, b,
                                                    (short)0, acc, false, false);
    }
    // C layout: VGPR v holds [M = v + 8*hv][N = lid]  -> transpose through LDS
    #pragma unroll
    for (int v = 0; v < 8; ++v)
        projL[wave][(v + 8 * hv) * 16 + lid] = acc[v];

    // ---- scale-weight softmax (tiny, per lane)
    float a0 = scw[0], a1 = scw[1], a2 = scw[2];
    float mxs = fmaxf(a0, fmaxf(a1, a2));
    float ee0 = __expf(a0 - mxs), ee1 = __expf(a1 - mxs), ee2 = __expf(a2 - mxs);
    float rsum = 1.0f / (ee0 + ee1 + ee2);
    float sw0 = ee0 * rsum, sw1 = ee1 * rsum, sw2 = ee2 * rsum;

    const long long t      = tok0 + lid;
    const long long OFF_W2 = (long long)ntok * D_MODEL;
    const long long OFF_W3 = OFF_W2 + (long long)ntok * 4;
    const long long OFF_W6 = OFF_W3 + (long long)ntok * 8;
    float* pr = &projL[wave][lid * 16];   // this lane's token row (M = lid)
    float* er = &eL[wave][lid * 16];

    // =========== vertex softmax: lanes 0-15 do k=2 & k=3; lanes 16-31 do k=6 (64 verts) ==========
    if (hv == 0) {
        // k = 2 (proj cols 0..1): V[v][j] = ((v >> (1-j)) & 1) ? +1 : -1
        float s0 = tanhf(pr[0]), s1 = tanhf(pr[1]);
        float ew2[4]; float sum2 = 0.f, n0 = 0.f, n1 = 0.f;
        #pragma unroll
        for (int v = 0; v < 4; ++v) {
            float c0 = (v & 2) ? 1.f : -1.f;
            float c1 = (v & 1) ? 1.f : -1.f;
            float e = __expf(c0 * s0 + c1 * s1);
            ew2[v] = e; sum2 += e; n0 += e * c0; n1 += e * c1;
        }
        float r2 = 1.f / sum2;
        #pragma unroll
        for (int v = 0; v < 4; ++v) out[OFF_W2 + t * 4 + v] = ew2[v] * r2;
        er[0] = sw0 * n0 * r2;
        er[1] = sw0 * n1 * r2;

        // k = 3 (proj cols 2..4)
        float u0 = tanhf(pr[2]), u1 = tanhf(pr[3]), u2 = tanhf(pr[4]);
        float ew3[8]; float sum3 = 0.f, m0 = 0.f, m1 = 0.f, m2 = 0.f;
        #pragma unroll
        for (int v = 0; v < 8; ++v) {
            float c0 = (v & 4) ? 1.f : -1.f;
            float c1 = (v & 2) ? 1.f : -1.f;
            float c2 = (v & 1) ? 1.f : -1.f;
            float e = __expf(c0 * u0 + c1 * u1 + c2 * u2);
            ew3[v] = e; sum3 += e; m0 += e * c0; m1 += e * c1; m2 += e * c2;
        }
        float r3 = 1.f / sum3;
        #pragma unroll
        for (int v = 0; v < 8; ++v) out[OFF_W3 + t * 8 + v] = ew3[v] * r3;
        er[2] = sw1 * m0 * r3;
        er[3] = sw1 * m1 * r3;
        er[4] = sw1 * m2 * r3;
    } else {
        // k = 6 (proj cols 5..10), 64 vertices; recompute exp on 2nd pass to cap VGPRs
        float s[6];
        #pragma unroll
        for (int j = 0; j < 6; ++j) s[j] = tanhf(pr[5 + j]);
        float sum6 = 0.f, num[6] = {0.f, 0.f, 0.f, 0.f, 0.f, 0.f};
        #pragma unroll
        for (int v = 0; v < 64; ++v) {
            float sim = 0.f;
            #pragma unroll
            for (int j = 0; j < 6; ++j)
                sim += ((v >> (5 - j)) & 1) ? s[j] : -s[j];
            float e = __expf(sim);
            sum6 += e;
            #pragma unroll
            for (int j = 0; j < 6; ++j)
                num[j] += ((v >> (5 - j)) & 1) ? e : -e;
        }
        float r6 = 1.f / sum6;
        #pragma unroll
        for (int v = 0; v < 64; ++v) {
            float sim = 0.f;
            #pragma unroll
            for (int j = 0; j < 6; ++j)
                sim += ((v >> (5 - j)) & 1) ? s[j] : -s[j];
            out[OFF_W6 + t * 64 + v] = __expf(sim) * r6;
        }
        #pragma unroll
        for (int j = 0; j < 6; ++j) er[5 + j] = sw2 * num[j] * r6;
        er[11] = 0.f;   // K pad 11->12
    }
    // (no barrier needed: producer/consumer lanes are in the same wave; LDS is in-order per wave)

    // =========== stage 2: y = x + e(16x12) @ Bo(12x2048), 3 WMMA per 16-col tile ==========
    // A operand per K-step s: lane needs e[m = lid][4s + 2*hv + {0,1}] (consecutive -> b64)
    v2f aw0 = *(const v2f*)(&eL[wave][lid * 16 + 0 + 2 * hv]);
    v2f aw1 = *(const v2f*)(&eL[wave][lid * 16 + 4 + 2 * hv]);
    v2f aw2 = *(const v2f*)(&eL[wave][lid * 16 + 8 + 2 * hv]);

    for (int nt = 0; nt < D_MODEL / 16; ++nt) {
        int col = nt * 16 + lid;
        v8f cc = {0.f, 0.f, 0.f, 0.f, 0.f, 0.f, 0.f, 0.f};
        // B K-pair index kk = 2s + hv  (pairs pre-interleaved in BoP)
        v2f b0 = *(const v2f*)(&BoP[2 * ((0 + hv) * D_MODEL + col)]);
        v2f b1 = *(const v2f*)(&BoP[2 * ((2 + hv) * D_MODEL + col)]);
        v2f b2 = *(const v2f*)(&BoP[2 * ((4 + hv) * D_MODEL + col)]);
        cc = __builtin_amdgcn_wmma_f32_16x16x4_f32(false, aw0, false, b0, (short)0, cc, false, false);
        cc = __builtin_amdgcn_wmma_f32_16x16x4_f32(false, aw1, false, b1, (short)0, cc, false, false);
        cc = __builtin_amdgcn_wmma_f32_16x16x4_f32(false, aw2, false, b2, (short)0, cc, false, false);
        // epilogue: residual add + store (x tile is L2-hot from stage 1)
        #pragma unroll
        for (int v = 0; v < 8; ++v) {
            long long idx = (tok0 + v + 8 * hv) * D_MODEL + nt * 16 + lid;
            out[idx] = x[idx] + cc[v];
        }
    }
}

extern "C" void kernel_launch(void* const* d_in, const int* in_sizes, int n_in,
                              void* d_out, int out_size, void* d_ws, size_t ws_size,
                              hipStream_t stream) {
    const float* x   = (const float*)d_in[0];
    const float* Wp2 = (const float*)d_in[1];
    const float* Wp3 = (const float*)d_in[2];
    const float* Wp6 = (const float*)d_in[3];
    const float* Wo2 = (const float*)d_in[4];
    const float* Wo3 = (const float*)d_in[5];
    const float* Wo6 = (const float*)d_in[6];
    const float* scw = (const float*)d_in[7];
    float* out = (float*)d_out;

    int ntok   = in_sizes[0] / D_MODEL;   // 8*4096 = 32768
    int blocks = ntok / BLK_TOK;          // 256
    msq6_fused<<<dim3(blocks), dim3(256), 0, stream>>>(
        x, Wp2, Wp3, Wp6, Wo2, Wo3, Wo6, scw, out, ntok);
}